// BatteryGNN_31241592111582
// MI455X (gfx1250) — compile-verified
//
#include <hip/hip_runtime.h>
#include <hip/hip_bf16.h>

typedef float v2f __attribute__((ext_vector_type(2)));
typedef float v8f __attribute__((ext_vector_type(8)));

#define N_NODES 20000
#define N_EDGES 160000
#define N_GRAPHS 256
#define HDIM 128
#define ZK 384            // 3*H
#define ZP 388            // padded LDS row stride (floats): 388 % 64 = 4 -> conflict-free
#define EPB 32            // edges per block (2 M-tiles, shared by both waves)
#define KSTEPS 96         // 384 / 4
#define PACK_PER_LAYER (8 * KSTEPS * 32 * 2)   // floats per packed weight layer

// ---------------- node / edge embeddings ----------------
__global__ __launch_bounds__(256) void gnn_node_embed(
    const float* __restrict__ x, const float* __restrict__ W,
    const float* __restrict__ b, float* __restrict__ h) {
  int i = blockIdx.x * blockDim.x + threadIdx.x;
  if (i >= N_NODES * HDIM) return;
  int row = i >> 7, ch = i & 127;
  float acc = b[ch];
#pragma unroll
  for (int k = 0; k < 10; ++k) acc += x[row * 10 + k] * W[k * HDIM + ch];
  h[i] = fmaxf(acc, 0.0f);
}

__global__ __launch_bounds__(256) void gnn_edge_embed(
    const float* __restrict__ ea, const float* __restrict__ W,
    const float* __restrict__ b, float* __restrict__ e) {
  int i = blockIdx.x * blockDim.x + threadIdx.x;
  if (i >= N_EDGES * HDIM) return;
  int row = i >> 7, ch = i & 127;
  float acc = b[ch];
#pragma unroll
  for (int k = 0; k < 3; ++k) acc += ea[row * 3 + k] * W[k * HDIM + ch];
  e[i] = fmaxf(acc, 0.0f);
}

// ---------------- weight pre-pack into WMMA fragment-major layout ----------
// out[((layer*8 + nt)*96 + kstep)*32 + lane] = {W[krow,n], W[krow+1,n]}
// with krow = 4*kstep + 2*(lane>=16), n = nt*16 + (lane&15).
__global__ __launch_bounds__(256) void gnn_pack_w(
    const float* __restrict__ W, float* __restrict__ out) {
  int i = blockIdx.x * blockDim.x + threadIdx.x;
  if (i >= 10 * 8 * KSTEPS * 32) return;
  int lane  = i & 31;
  int kstep = (i >> 5) % KSTEPS;
  int nt    = ((i >> 5) / KSTEPS) & 7;
  int layer = i / (8 * KSTEPS * 32);
  int half = lane >> 4, l16 = lane & 15;
  int krow = kstep * 4 + 2 * half;
  int n = nt * 16 + l16;
  const float* Wl = W + (size_t)layer * ZK * HDIM;
  float2 v;
  v.x = Wl[(size_t)krow * HDIM + n];
  v.y = Wl[(size_t)(krow + 1) * HDIM + n];
  ((float2*)out)[i] = v;
}

// ---------------- CGConv message kernel: fp32 WMMA 16x16x4 ----------------
// z = [h[dst] | h[src] | e]  (E x 384), msg = sigmoid(z@Wf+bf)*softplus(z@Ws+bs)
// h_out (pre-loaded with residual h_in) += segment_sum(msg, dst) via f32 atomics.
// Block = 64 threads (2 waves) staging 32 edges. Wave w owns N-tiles
// [4w, 4w+4) and computes BOTH 16-edge M-tiles, so each packed B-fragment
// (one global_load_b64 per lane per matrix) feeds 2 WMMAs, one per M-tile.
__global__ __launch_bounds__(64) void cg_edge_msg(
    const float* __restrict__ h_in, const float* __restrict__ e_feat,
    const int* __restrict__ srcI, const int* __restrict__ dstI,
    const float* __restrict__ Pf, const float* __restrict__ bfp,
    const float* __restrict__ Ps, const float* __restrict__ bsp,
    float* __restrict__ h_out) {
  __shared__ float zsm[EPB * ZP];   // 32 * 388 * 4B = 49.7 KB
  __shared__ int dsm[EPB];

  const int e0 = blockIdx.x * EPB;
  // ---- gather phase: 32 rows x 96 float4 each, 64 threads ----
  for (int idx = threadIdx.x; idx < EPB * 96; idx += 64) {
    int r = idx / 96;
    int c4 = idx - r * 96;
    int eid = e0 + r;
    if (eid >= N_EDGES) eid = N_EDGES - 1;
    int sN = srcI[eid];
    int dN = dstI[eid];
    float4 val;
    if (c4 < 32)      val = ((const float4*)(h_in  + (size_t)dN  * HDIM))[c4];
    else if (c4 < 64) val = ((const float4*)(h_in  + (size_t)sN  * HDIM))[c4 - 32];
    else              val = ((const float4*)(e_feat + (size_t)eid * HDIM))[c4 - 64];
    *(float4*)&zsm[r * ZP + c4 * 4] = val;
    if (c4 == 0) dsm[r] = dN;
  }
  __syncthreads();

  // ---- WMMA phase (full EXEC) ----
  const int lane = threadIdx.x & 31;
  const int wv   = threadIdx.x >> 5;
  const int l16  = lane & 15;
  const int half = lane >> 4;                     // 0: lanes 0-15, 1: lanes 16-31
  const float* z0 = &zsm[l16 * ZP];               // M-tile 0: edge rows 0-15
  const float* z1 = &zsm[(16 + l16) * ZP];        // M-tile 1: edge rows 16-31

  for (int j = 0; j < 4; ++j) {
    const int nt = wv * 4 + j;
    const int n = nt * 16 + l16;                  // output channel (B/C/D col)
    const float2* pf = (const float2*)Pf + (size_t)nt * KSTEPS * 32 + lane;
    const float2* ps = (const float2*)Ps + (size_t)nt * KSTEPS * 32 + lane;
    v8f aF0 = {0.f, 0.f, 0.f, 0.f, 0.f, 0.f, 0.f, 0.f};
    v8f aS0 = aF0, aF1 = aF0, aS1 = aF0;
    for (int ks = 0; ks < KSTEPS; ++ks) {
      const int krow = ks * 4 + 2 * half;
      v2f a0, a1;                                 // A 16x4 f32: v0=K(+0/+2), v1=K+1
      a0.x = z0[krow];  a0.y = z0[krow + 1];
      a1.x = z1[krow];  a1.y = z1[krow + 1];
      float2 wf = pf[ks * 32];                    // one b64 per lane per matrix
      float2 wsv = ps[ks * 32];
      v2f bF, bS;
      bF.x = wf.x;  bF.y = wf.y;
      bS.x = wsv.x; bS.y = wsv.y;
      aF0 = __builtin_amdgcn_wmma_f32_16x16x4_f32(false, a0, false, bF,
                                                  (short)0, aF0, false, false);
      aS0 = __builtin_amdgcn_wmma_f32_16x16x4_f32(false, a0, false, bS,
                                                  (short)0, aS0, false, false);
      aF1 = __builtin_amdgcn_wmma_f32_16x16x4_f32(false, a1, false, bF,
                                                  (short)0, aF1, false, false);
      aS1 = __builtin_amdgcn_wmma_f32_16x16x4_f32(false, a1, false, bS,
                                                  (short)0, aS1, false, false);
    }
    const float bFc = bfp[n];
    const float bSc = bsp[n];
#pragma unroll
    for (int r = 0; r < 8; ++r) {                 // C/D rows: r (half 0) / r+8 (half 1)
      const int m0 = r + 8 * half;
      const int m1 = 16 + m0;
      {
        float f = aF0[r] + bFc, s = aS0[r] + bSc;
        float sig = 1.0f / (1.0f + __expf(-f));
        float sp  = fmaxf(s, 0.0f) + log1pf(__expf(-fabsf(s)));
        if (e0 + m0 < N_EDGES)
          atomicAdd(&h_out[(size_t)dsm[m0] * HDIM + n], sig * sp);
      }
      {
        float f = aF1[r] + bFc, s = aS1[r] + bSc;
        float sig = 1.0f / (1.0f + __expf(-f));
        float sp  = fmaxf(s, 0.0f) + log1pf(__expf(-fabsf(s)));
        if (e0 + m1 < N_EDGES)
          atomicAdd(&h_out[(size_t)dsm[m1] * HDIM + n], sig * sp);
      }
    }
  }
}

// ---------------- BN(eval) + ReLU (+ alternating residual) ----------------
__global__ __launch_bounds__(256) void gnn_bn_relu(
    float* __restrict__ h, const float* __restrict__ hprev,
    const float* __restrict__ g, const float* __restrict__ b,
    const float* __restrict__ m, const float* __restrict__ v, int add_res) {
  int i = blockIdx.x * blockDim.x + threadIdx.x;
  if (i >= N_NODES * HDIM) return;
  int ch = i & 127;
  float val = (h[i] - m[ch]) * (g[ch] * rsqrtf(v[ch] + 1e-5f)) + b[ch];
  val = fmaxf(val, 0.0f);
  if (add_res) val += hprev[i];
  h[i] = val;
}

// ---------------- pooling (sum / max / count) ----------------
__global__ __launch_bounds__(256) void gnn_pool(
    const float* __restrict__ h, const int* __restrict__ batch,
    float* __restrict__ psum, float* __restrict__ pmax, float* __restrict__ pcnt) {
  int i = blockIdx.x * blockDim.x + threadIdx.x;
  if (i >= N_NODES * HDIM) return;
  int node = i >> 7, ch = i & 127;
  int g = batch[node];
  float v = h[i];
  atomicAdd(&psum[g * HDIM + ch], v);
  atomicMax((int*)&pmax[g * HDIM + ch], __float_as_int(v));  // v >= 0 post-ReLU
  if (ch == 0) atomicAdd(&pcnt[g], 1.0f);
}

// ---------------- graph MLP head ----------------
__global__ __launch_bounds__(256) void gnn_head1(
    const float* __restrict__ psum, const float* __restrict__ pmax,
    const float* __restrict__ pcnt, const float* __restrict__ W1,
    const float* __restrict__ b1, const float* __restrict__ g_,
    const float* __restrict__ b_, const float* __restrict__ m_,
    const float* __restrict__ v_, float* __restrict__ out) {
  int i = blockIdx.x * blockDim.x + threadIdx.x;
  if (i >= N_GRAPHS * 256) return;
  int gg = i >> 8, oc = i & 255;
  float cnt = fmaxf(pcnt[gg], 1.0f);
  float acc = b1[oc];
  for (int k = 0; k < HDIM; ++k) {
    float s = psum[gg * HDIM + k];
    acc += (s / cnt) * W1[(size_t)k * 256 + oc];
    acc += pmax[gg * HDIM + k] * W1[(size_t)(HDIM + k) * 256 + oc];
    acc += s * W1[(size_t)(2 * HDIM + k) * 256 + oc];
  }
  acc = (acc - m_[oc]) * (g_[oc] * rsqrtf(v_[oc] + 1e-5f)) + b_[oc];
  out[i] = fmaxf(acc, 0.0f);
}

__global__ __launch_bounds__(256) void gnn_head2(
    const float* __restrict__ g1, const float* __restrict__ W2,
    const float* __restrict__ b2, const float* __restrict__ g_,
    const float* __restrict__ b_, const float* __restrict__ m_,
    const float* __restrict__ v_, float* __restrict__ out) {
  int i = blockIdx.x * blockDim.x + threadIdx.x;
  if (i >= N_GRAPHS * 128) return;
  int gg = i >> 7, oc = i & 127;
  float acc = b2[oc];
  for (int k = 0; k < 256; ++k) acc += g1[gg * 256 + k] * W2[(size_t)k * 128 + oc];
  acc = (acc - m_[oc]) * (g_[oc] * rsqrtf(v_[oc] + 1e-5f)) + b_[oc];
  out[i] = fmaxf(acc, 0.0f);
}

__global__ __launch_bounds__(256) void gnn_head3(
    const float* __restrict__ g2, const float* __restrict__ W3,
    const float* __restrict__ b3, float* __restrict__ out) {
  int i = blockIdx.x * blockDim.x + threadIdx.x;
  if (i >= N_GRAPHS * 64) return;
  int gg = i >> 6, oc = i & 63;
  float acc = b3[oc];
  for (int k = 0; k < 128; ++k) acc += g2[gg * 128 + k] * W3[(size_t)k * 64 + oc];
  out[i] = fmaxf(acc, 0.0f);
}

__global__ __launch_bounds__(256) void gnn_head4(
    const float* __restrict__ g3,
    const float* __restrict__ Wv, const float* __restrict__ bv,
    const float* __restrict__ Wen, const float* __restrict__ ben,
    const float* __restrict__ Wd, const float* __restrict__ bd,
    const float* __restrict__ Wh, const float* __restrict__ bh,
    float* __restrict__ out) {
  int i = blockIdx.x * blockDim.x + threadIdx.x;
  if (i >= N_GRAPHS * 4) return;
  int gg = i & 255, hh = i >> 8;
  const float* W = (hh == 0) ? Wv : (hh == 1) ? Wen : (hh == 2) ? Wd : Wh;
  float acc = ((hh == 0) ? bv : (hh == 1) ? ben : (hh == 2) ? bd : bh)[0];
  for (int k = 0; k < 64; ++k) acc += g3[gg * 64 + k] * W[k];
  out[hh * N_GRAPHS + gg] = acc;
}

// ---------------- launcher ----------------
extern "C" void kernel_launch(void* const* d_in, const int* in_sizes, int n_in,
                              void* d_out, int out_size, void* d_ws, size_t ws_size,
                              hipStream_t stream) {
  const float* x      = (const float*)d_in[0];
  const float* eattr  = (const float*)d_in[1];
  const int*   eidx   = (const int*)  d_in[2];
  const int*   batch  = (const int*)  d_in[3];
  const float* W_node = (const float*)d_in[4];
  const float* b_node = (const float*)d_in[5];
  const float* W_edge = (const float*)d_in[6];
  const float* b_edge = (const float*)d_in[7];
  const float* Wf     = (const float*)d_in[8];
  const float* bf     = (const float*)d_in[9];
  const float* Ws     = (const float*)d_in[10];
  const float* bs     = (const float*)d_in[11];
  const float* bng    = (const float*)d_in[12];
  const float* bnb    = (const float*)d_in[13];
  const float* bnm    = (const float*)d_in[14];
  const float* bnv    = (const float*)d_in[15];
  const float* W1     = (const float*)d_in[16];
  const float* b1     = (const float*)d_in[17];
  const float* bn1g   = (const float*)d_in[18];
  const float* bn1b   = (const float*)d_in[19];
  const float* bn1m   = (const float*)d_in[20];
  const float* bn1v   = (const float*)d_in[21];
  const float* W2     = (const float*)d_in[22];
  const float* b2     = (const float*)d_in[23];
  const float* bn2g   = (const float*)d_in[24];
  const float* bn2b   = (const float*)d_in[25];
  const float* bn2m   = (const float*)d_in[26];
  const float* bn2v   = (const float*)d_in[27];
  const float* W3     = (const float*)d_in[28];
  const float* b3     = (const float*)d_in[29];
  const float* Wv     = (const float*)d_in[30];
  const float* bv     = (const float*)d_in[31];
  const float* Wen    = (const float*)d_in[32];
  const float* ben    = (const float*)d_in[33];
  const float* Wd     = (const float*)d_in[34];
  const float* bd     = (const float*)d_in[35];
  const float* Wh     = (const float*)d_in[36];
  const float* bh     = (const float*)d_in[37];

  // workspace layout (floats)
  float* ws   = (float*)d_ws;
  float* hA   = ws;
  float* hB   = hA + (size_t)N_NODES * HDIM;
  float* ef   = hB + (size_t)N_NODES * HDIM;
  float* psum = ef + (size_t)N_EDGES * HDIM;
  float* pmax = psum + (size_t)N_GRAPHS * HDIM;
  float* pcnt = pmax + (size_t)N_GRAPHS * HDIM;
  float* g1   = pcnt + N_GRAPHS;
  float* g2   = g1 + (size_t)N_GRAPHS * 256;
  float* g3   = g2 + (size_t)N_GRAPHS * 128;
  float* pWf  = g3 + (size_t)N_GRAPHS * 64;
  float* pWs  = pWf + (size_t)10 * PACK_PER_LAYER;

  const int* srcI = eidx;             // edge_index[0] = src
  const int* dstI = eidx + N_EDGES;   // edge_index[1] = dst

  gnn_node_embed<<<(N_NODES * HDIM + 255) / 256, 256, 0, stream>>>(x, W_node, b_node, hA);
  gnn_edge_embed<<<(N_EDGES * HDIM + 255) / 256, 256, 0, stream>>>(eattr, W_edge, b_edge, ef);

  const int packThreads = 10 * 8 * KSTEPS * 32;
  gnn_pack_w<<<(packThreads + 255) / 256, 256, 0, stream>>>(Wf, pWf);
  gnn_pack_w<<<(packThreads + 255) / 256, 256, 0, stream>>>(Ws, pWs);

  float* hcur = hA;
  float* hnext = hB;
  for (int i = 0; i < 10; ++i) {
    // residual base: h_out starts as h_in, atomics accumulate messages into it
    hipMemcpyAsync(hnext, hcur, (size_t)N_NODES * HDIM * sizeof(float),
                   hipMemcpyDeviceToDevice, stream);
    cg_edge_msg<<<N_EDGES / EPB, 64, 0, stream>>>(
        hcur, ef, srcI, dstI,
        pWf + (size_t)i * PACK_PER_LAYER, bf + (size_t)i * HDIM,
        pWs + (size_t)i * PACK_PER_LAYER, bs + (size_t)i * HDIM, hnext);
    gnn_bn_relu<<<(N_NODES * HDIM + 255) / 256, 256, 0, stream>>>(
        hnext, hcur, bng + (size_t)i * HDIM, bnb + (size_t)i * HDIM,
        bnm + (size_t)i * HDIM, bnv + (size_t)i * HDIM, (i & 1));
    float* t = hcur; hcur = hnext; hnext = t;
  }

  hipMemsetAsync(psum, 0, (size_t)(2 * N_GRAPHS * HDIM + N_GRAPHS) * sizeof(float), stream);
  gnn_pool<<<(N_NODES * HDIM + 255) / 256, 256, 0, stream>>>(hcur, batch, psum, pmax, pcnt);

  gnn_head1<<<(N_GRAPHS * 256 + 255) / 256, 256, 0, stream>>>(
      psum, pmax, pcnt, W1, b1, bn1g, bn1b, bn1m, bn1v, g1);
  gnn_head2<<<(N_GRAPHS * 128 + 255) / 256, 256, 0, stream>>>(
      g1, W2, b2, bn2g, bn2b, bn2m, bn2v, g2);
  gnn_head3<<<(N_GRAPHS * 64 + 255) / 256, 256, 0, stream>>>(g2, W3, b3, g3);
  gnn_head4<<<(N_GRAPHS * 4 + 255) / 256, 256, 0, stream>>>(
      g3, Wv, bv, Wen, ben, Wd, bd, Wh, bh, (float*)d_out);
}